// StridedSlice_28767690948752
// MI455X (gfx1250) — compile-verified
//
#include <hip/hip_runtime.h>

// Reference reduces to: out[n][c][ho][wo] = x[n][c][2*ho][2*wo]
//   x:(8,128,512,512) f32  ->  out:(8,128,256,256) f32
// Pure data movement. Traffic floor: 512MB read (even input rows only; W-stride-2
// cannot skip below cacheline granularity) + 256MB write = 768MB -> ~33us @ 23.3TB/s.
//
// CDNA5 path: double-buffered GLOBAL_LOAD_ASYNC_TO_LDS_B128 per wave (ASYNCcnt),
// conflict-free strided compaction out of LDS, coalesced 128B-per-instruction stores.
// Per-wave-private LDS buffers -> zero barriers; only split wait counters.

#define N_  8
#define C_  128
#define H_  512
#define W_  512
#define HO_ 256
#define WO_ 256

constexpr int ROWS            = N_ * C_ * HO_;            // 262144 output rows
constexpr int IN_PLANE        = H_ * W_;                  // floats per (n,c) plane
constexpr int WAVES_PER_BLOCK = 8;                        // 256 threads, wave32
constexpr int BLOCKS          = 2048;
constexpr int TOTAL_WAVES     = BLOCKS * WAVES_PER_BLOCK; // 16384
constexpr int ITERS           = ROWS / TOTAL_WAVES;       // 16 (exact)
constexpr int LDS_ROW         = W_;                       // 512 floats staged per row

// Builtin signature (from clang diagnostic): (int4 AS1*, int4 AS3*, imm, imm)
typedef int v4i __attribute__((vector_size(16)));
typedef __attribute__((address_space(1))) v4i GV4;        // global int4
typedef __attribute__((address_space(3))) v4i LV4;        // LDS int4

__device__ __forceinline__ void async_row_load(const float* gsrc, float* ldst) {
  // 32 lanes x 64B = one full 2KB input row; 16B-aligned on both sides.
  GV4* g = (GV4*)gsrc;
  LV4* l = (LV4*)ldst;
  __builtin_amdgcn_global_load_async_to_lds_b128(g, l,  0, 0);
  __builtin_amdgcn_global_load_async_to_lds_b128(g, l, 16, 0);
  __builtin_amdgcn_global_load_async_to_lds_b128(g, l, 32, 0);
  __builtin_amdgcn_global_load_async_to_lds_b128(g, l, 48, 0);
}

__global__ __launch_bounds__(256) void strided_slice_kernel(
    const float* __restrict__ x, float* __restrict__ out) {
  __shared__ float lds[WAVES_PER_BLOCK * 2 * LDS_ROW];    // 32 KB, per-wave double buffers

  const int lane  = threadIdx.x & 31;
  const int wave  = threadIdx.x >> 5;
  const int gwave = blockIdx.x * WAVES_PER_BLOCK + wave;

  float* wlds = &lds[wave * 2 * LDS_ROW];

  auto in_row_ptr = [&](int row) -> const float* {
    const int plane = row >> 8;          // 256 output rows per (n,c) plane
    const int ho    = row & (HO_ - 1);
    // lane covers input floats [16L, 16L+16) of input row 2*ho
    return x + (size_t)plane * IN_PLANE + (size_t)(ho << 1) * W_ + lane * 16;
  };

  // Prologue: kick off row for iteration 0 into buffer 0.
  async_row_load(in_row_ptr(gwave), wlds + lane * 16);

  for (int t = 0; t < ITERS; ++t) {
    const int row = gwave + t * TOTAL_WAVES;

    if (t + 1 < ITERS) {
      // DS loads vs later ASYNC writes to the same LDS buffer are unordered
      // across counters: ensure previous compaction's ds reads fully drained
      // before the DMA is allowed to overwrite that buffer.
      asm volatile("s_wait_dscnt 0x0" ::: "memory");
      async_row_load(in_row_ptr(row + TOTAL_WAVES),
                     wlds + ((t + 1) & 1) * LDS_ROW + lane * 16);
      // 4 newest async loads (next row) may remain in flight; the 4 older ones
      // (this row) are complete once ASYNCcnt <= 4 (in-order completion).
      asm volatile("s_wait_asynccnt 0x4" ::: "memory");
    } else {
      asm volatile("s_wait_asynccnt 0x0" ::: "memory");
    }

    // Compaction: output float j = L + 32k  <-  LDS float 2L + 64k.
    // LDS bank = (2L + 64k) % 64 = 2L : conflict-free across wave32.
    // Each store instruction writes a contiguous 128B across the wave.
    const float* lsrc = wlds + (t & 1) * LDS_ROW;
    float* orow = out + (size_t)row * WO_ + lane;
#pragma unroll
    for (int k = 0; k < 8; ++k) {
      orow[32 * k] = lsrc[2 * lane + 64 * k];
    }
  }
}

extern "C" void kernel_launch(void* const* d_in, const int* in_sizes, int n_in,
                              void* d_out, int out_size, void* d_ws, size_t ws_size,
                              hipStream_t stream) {
  const float* x  = (const float*)d_in[0];   // (8,128,512,512) f32
  float* out      = (float*)d_out;           // (8,128,256,256) f32
  (void)in_sizes; (void)n_in; (void)out_size; (void)d_ws; (void)ws_size;
  strided_slice_kernel<<<BLOCKS, 256, 0, stream>>>(x, out);
}